// LearnedGraphAttn_10771777978707
// MI455X (gfx1250) — compile-verified
//
#include <hip/hip_runtime.h>
#include <hip/hip_bf16.h>

// Problem constants (match reference)
#define BB    4
#define NN    4096
#define FIN   128
#define HID   64
#define TOPKK 32
#define ROWT  16            // rows per workgroup in k_proj / k_attn_rows
#define NTIL  (NN / 16)     // 256 column tiles of 16
#define STILE 64            // symmetrize tile
#define MASKFILL (-1e9f)

typedef __attribute__((ext_vector_type(2))) float v2f;
typedef __attribute__((ext_vector_type(8))) float v8f;

__device__ __forceinline__ v8f wmma_f32_16x16x4(v2f a, v2f b, v8f c) {
  // D = A(16x4,f32) x B(4x16,f32) + C(16x16,f32) -- exact f32 WMMA path
  return __builtin_amdgcn_wmma_f32_16x16x4_f32(false, a, false, b, (short)0, c,
                                               false, false);
}

// ---------------------------------------------------------------------------
// Kernel 0: zero the column-sum accumulator
// ---------------------------------------------------------------------------
__global__ void k_zero(float* __restrict__ p, int n) {
  int i = blockIdx.x * blockDim.x + threadIdx.x;
  if (i < n) p[i] = 0.0f;
}

// ---------------------------------------------------------------------------
// Kernel 1: Z = H @ W + b   (B*N x 128) @ (128 x 64)
// grid = B*N/16 blocks, 128 threads (4 waves); wave w owns output cols 16w..16w+15
// ---------------------------------------------------------------------------
__global__ __launch_bounds__(128) void k_proj(const float* __restrict__ H,
                                              const float* __restrict__ W,
                                              const float* __restrict__ bias,
                                              float* __restrict__ Z) {
  const int lane    = threadIdx.x & 31;
  const int wave    = threadIdx.x >> 5;        // 0..3
  const int rowBase = blockIdx.x * ROWT;       // flat row in [0, B*N)
  const int m       = lane & 15;               // A-matrix row / B-matrix col
  const int koff    = (lane >> 4) << 1;        // 0 for lanes 0-15, 2 for 16-31
  const int hi      = (lane >> 4) * 8;         // D row offset
  const int colBase = wave * 16;

  const float* hrow = H + (size_t)(rowBase + m) * FIN;
  v8f acc = {};
  #pragma unroll 8
  for (int kk = 0; kk < FIN; kk += 4) {
    v2f a, b;
    a.x = hrow[kk + koff];
    a.y = hrow[kk + koff + 1];
    b.x = W[(size_t)(kk + koff)     * HID + colBase + m];
    b.y = W[(size_t)(kk + koff + 1) * HID + colBase + m];
    acc = wmma_f32_16x16x4(a, b, acc);
  }
  const float bv = bias[colBase + m];
  #pragma unroll
  for (int v = 0; v < 8; ++v) {
    const int r = hi + v;
    Z[(size_t)(rowBase + r) * HID + colBase + m] = acc[v] + bv;
  }
}

// ---------------------------------------------------------------------------
// Kernel 2: per 16-row strip: masked scaled logits (WMMA, full strip in LDS),
// top-32 per row, softmax, dense write to out, colsum atomics.
// grid = B*N/16 blocks, 256 threads (8 waves), dynamic LDS = 16*4096*4 = 256KB
// ---------------------------------------------------------------------------
__global__ __launch_bounds__(256) void k_attn_rows(const float* __restrict__ Z,
                                                   const unsigned char* __restrict__ mask,
                                                   float* __restrict__ out,
                                                   float* __restrict__ colsum) {
  extern __shared__ float lds[];               // ROWT * NN floats (256 KB)
  const int lane        = threadIdx.x & 31;
  const int wave        = threadIdx.x >> 5;    // 0..7
  const int flatRowBase = blockIdx.x * ROWT;   // = b*N + rowInB
  const int b           = flatRowBase / NN;
  const int rowInB      = flatRowBase - b * NN;
  const int m           = lane & 15;
  const int koff        = (lane >> 4) << 1;
  const int hi          = (lane >> 4) * 8;

  // ---- logits strip: 16 x 4096, 16x16 tiles via V_WMMA_F32_16X16X4_F32 ----
  const float* arow = Z + (size_t)(flatRowBase + m) * HID;
  for (int t = wave; t < NTIL; t += 8) {
    const int colB = t * 16;
    const float* brow = Z + (size_t)(b * NN + colB + m) * HID;  // B = Zc^T
    v8f acc = {};
    #pragma unroll
    for (int kk = 0; kk < HID; kk += 4) {
      v2f a, bm;
      a.x  = arow[kk + koff];  a.y  = arow[kk + koff + 1];
      bm.x = brow[kk + koff];  bm.y = brow[kk + koff + 1];
      acc = wmma_f32_16x16x4(a, bm, acc);
    }
    #pragma unroll
    for (int v = 0; v < 8; ++v) {
      const int r  = hi + v;                   // local row 0..15
      const int gc = colB + m;                 // global column
      const unsigned char mk = mask[(size_t)(rowInB + r) * NN + gc];
      lds[r * NN + gc] = mk ? acc[v] * 0.125f : MASKFILL;  // /sqrt(64)
    }
  }
  __syncthreads();

  // ---- per-row top-32 + softmax: wave handles rows 2w and 2w+1 ----
  for (int rr = 0; rr < 2; ++rr) {
    const int row = wave * 2 + rr;
    float* rowp = lds + row * NN;
    float firstMax = 0.0f;
    float myE = 0.0f;
    int   myIdx = lane;                        // distinct defaults

    for (int k = 0; k < TOPKK; ++k) {
      float bv = -__builtin_inff();
      int   bi = 0x7fffffff;
      for (int j = lane; j < NN; j += 32) {
        const float v = rowp[j];
        if (v > bv) { bv = v; bi = j; }
      }
      #pragma unroll
      for (int off = 16; off > 0; off >>= 1) {  // argmax, tie -> lower index
        const float ov = __shfl_xor(bv, off, 32);
        const int   oi = __shfl_xor(bi, off, 32);
        if (ov > bv || (ov == bv && oi < bi)) { bv = ov; bi = oi; }
      }
      if (k == 0) firstMax = bv;               // uniform across wave
      if (lane == k) { myE = __expf(bv - firstMax); myIdx = bi; }
      if (lane == 0) rowp[bi] = -__builtin_inff();  // LDS in-order per wave
    }
    // softmax denominator (lane k holds k-th exp term)
    float s = myE;
    #pragma unroll
    for (int off = 16; off > 0; off >>= 1) s += __shfl_xor(s, off, 32);
    const float p = myE / s;                   // s >= exp(0) = 1

    // dense row: zeros + scatter in LDS, stream to global, colsum atomics
    for (int j = lane; j < NN; j += 32) rowp[j] = 0.0f;
    rowp[myIdx] = p;                           // 32 distinct indices
    float* orow = out + (size_t)(flatRowBase + row) * NN;
    for (int j = lane; j < NN; j += 32) orow[j] = rowp[j];
    atomicAdd(&colsum[b * NN + myIdx], p);
  }
}

// ---------------------------------------------------------------------------
// Kernel 3: in-place symmetrize + row renorm using rowsum(S)=0.5*(1+colsum).
// grid = (64, 64, B); keeps only I<=J tile pairs; LDS staging avoids races.
// ---------------------------------------------------------------------------
__global__ __launch_bounds__(256) void k_sym(float* __restrict__ out,
                                             const float* __restrict__ colsum) {
  const int I = blockIdx.x, J = blockIdx.y, b = blockIdx.z;
  if (J < I) return;
  __shared__ float T1[STILE][STILE + 1];
  __shared__ float T2[STILE][STILE + 1];
  __shared__ float rI[STILE], rJ[STILE];
  const int tid = threadIdx.x;
  float* base = out + (size_t)b * NN * NN;

  for (int e = tid; e < STILE * STILE; e += 256) {
    const int i = e >> 6, j = e & 63;
    T1[i][j] = base[(size_t)(I * STILE + i) * NN + (J * STILE + j)];
  }
  if (J != I) {
    for (int e = tid; e < STILE * STILE; e += 256) {
      const int i = e >> 6, j = e & 63;
      T2[i][j] = base[(size_t)(J * STILE + i) * NN + (I * STILE + j)];
    }
  }
  if (tid < STILE) {
    const float r = 0.5f * (1.0f + colsum[b * NN + I * STILE + tid]);
    rI[tid] = 1.0f / fmaxf(r, 1e-12f);
  } else if (tid < 2 * STILE) {
    const int t = tid - STILE;
    const float r = 0.5f * (1.0f + colsum[b * NN + J * STILE + t]);
    rJ[t] = 1.0f / fmaxf(r, 1e-12f);
  }
  __syncthreads();

  if (I == J) {
    for (int e = tid; e < STILE * STILE; e += 256) {
      const int i = e >> 6, j = e & 63;
      const float v = 0.5f * (T1[i][j] + T1[j][i]);
      base[(size_t)(I * STILE + i) * NN + (J * STILE + j)] = v * rI[i];
    }
  } else {
    for (int e = tid; e < STILE * STILE; e += 256) {
      const int i = e >> 6, j = e & 63;
      const float v = 0.5f * (T1[i][j] + T2[j][i]);
      base[(size_t)(I * STILE + i) * NN + (J * STILE + j)] = v * rI[i];
      base[(size_t)(J * STILE + j) * NN + (I * STILE + i)] = v * rJ[j];
    }
  }
}

// ---------------------------------------------------------------------------
extern "C" void kernel_launch(void* const* d_in, const int* in_sizes, int n_in,
                              void* d_out, int out_size, void* d_ws, size_t ws_size,
                              hipStream_t stream) {
  (void)in_sizes; (void)n_in; (void)out_size; (void)ws_size;
  const float*         H    = (const float*)d_in[0];          // (B,N,128) f32
  const float*         W    = (const float*)d_in[1];          // (128,64)  f32
  const float*         bias = (const float*)d_in[2];          // (64,)     f32
  const unsigned char* mask = (const unsigned char*)d_in[3];  // (N,N) bool (1B)
  float* out = (float*)d_out;                                 // (B,N,N) f32

  float* Z      = (float*)d_ws;                // B*N*HID floats (4 MB)
  float* csum   = Z + (size_t)BB * NN * HID;   // B*N floats (64 KB)

  k_zero<<<(BB * NN + 255) / 256, 256, 0, stream>>>(csum, BB * NN);
  k_proj<<<BB * NN / ROWT, 128, 0, stream>>>(H, W, bias, Z);
  k_attn_rows<<<BB * NN / ROWT, 256, ROWT * NN * sizeof(float), stream>>>(
      Z, mask, out, csum);
  dim3 g3(NN / STILE, NN / STILE, BB);
  k_sym<<<g3, 256, 0, stream>>>(out, csum);
}